// SIPSEModel_61203283968724
// MI455X (gfx1250) — compile-verified
//
#include <hip/hip_runtime.h>
#include <stddef.h>

// ---------------------------------------------------------------------------
// SIPSE / GCN pipeline for MI455X (gfx1250, wave32, WMMA)
//
//   h1 = relu(Agg(x  @ W1) + b1)   256 -> 256
//   h2 = relu(Agg(h1 @ W2) + b2)   256 -> 128
//   h3 = relu(Agg(h2 @ W3) + b3)   128 -> 32
//   z  = relu(Agg(h3 @ Wkp) + bkp)  32 -> 16 (zero-padded from 8)
//   out[k,e] = dot8(z[src_e], z[dst_e])  replicated over k (K slices identical)
//
// GEMMs: V_WMMA_F32_16X16X4_F32, 5 M-tiles per wave, B tile staged in LDS,
// all strides compile-time. Aggregation: global f32 atomics (L2-resident).
// ---------------------------------------------------------------------------

typedef __attribute__((ext_vector_type(2))) float v2f;
typedef __attribute__((ext_vector_type(8))) float v8f;

#define N_NODES 50000      // = 3125 * 16 = 125 * 25 * 16
#define F_IN    256
#define MT      5          // M-tiles per wave
#define WPB     5          // waves per block (160 threads)
#define MG      (MT * WPB) // 25 M-tiles per block; 3125/25 = 125 blocks/column

// ---------------- utility kernels ----------------

__global__ void fill_f32_kernel(float* __restrict__ p, long long n, float v) {
    long long i = (long long)blockIdx.x * blockDim.x + threadIdx.x;
    if (i < n) p[i] = v;
}

__global__ void deg_count_kernel(const int* __restrict__ dst,
                                 float* __restrict__ deg, int E) {
    int e = blockIdx.x * blockDim.x + threadIdx.x;
    if (e < E) atomicAdd(deg + dst[e], 1.0f);
}

__global__ void rsqrt_kernel(float* __restrict__ deg, int n) {
    int i = blockIdx.x * blockDim.x + threadIdx.x;
    if (i < n) deg[i] = rsqrtf(fmaxf(deg[i], 1.0f));
}

// Zero-pad Wk [32 x 8] -> Wkp [32 x 16], bk [8] -> bkp [16].
__global__ void pad_wk_kernel(const float* __restrict__ Wk,
                              const float* __restrict__ bk,
                              float* __restrict__ Wkp,
                              float* __restrict__ bkp) {
    int i = blockIdx.x * blockDim.x + threadIdx.x;
    if (i < 32 * 16) {
        const int r = i >> 4, c = i & 15;
        Wkp[i] = (c < 8) ? Wk[r * 8 + c] : 0.0f;
    }
    if (i < 16) bkp[i] = (i < 8) ? bk[i] : 0.0f;
}

// ---------------- fp32 WMMA GEMM:  C[M x NW] = A[M x K] @ B[K x NW] --------
// K, NW compile-time; M = 50000 fixed by grid. One block = 5 waves, each wave
// computes 5 16x16 tiles in one N column; B column tile staged in LDS.
// fp32 16x16x4 striping (ISA 7.12.2):
//   A (16x4): lane {half=lane>>4, m=lane&15}; vgpr j = A[m][2*half+j]
//   B (4x16): vgpr j = B[2*half+j][n], n = lane&15
//   C (16x16): vgpr r = C[r + 8*half][n]
template <int K, int NW>
__global__ __launch_bounds__(160)
void gemm_wmma_kernel(const float* __restrict__ A,
                      const float* __restrict__ B,
                      float* __restrict__ C) {
    constexpr int tilesN = NW / 16;
    __shared__ float ldsB[K * 16];

    const int nt = (int)(blockIdx.x % tilesN);
    const int mg = (int)(blockIdx.x / tilesN);

    // cooperative stage of B tile [K x 16] (<= 16 KB LDS)
    for (int i = threadIdx.x; i < K * 16; i += 160) {
        const int kr = i >> 4, c = i & 15;
        ldsB[i] = B[(size_t)kr * NW + nt * 16 + c];
    }
    __syncthreads();

    const int wave = threadIdx.x >> 5;
    const int lane = threadIdx.x & 31;
    const int half = lane >> 4;
    const int l16  = lane & 15;

    const int tm0 = mg * MG + wave * MT;      // first of MT consecutive tiles
    const int col = nt * 16 + l16;

    const float* __restrict__ a0 =
        A + (size_t)(tm0 * 16 + l16) * K + (half << 1);

    const v8f vzero = {0.f, 0.f, 0.f, 0.f, 0.f, 0.f, 0.f, 0.f};
    v8f acc[MT];
    #pragma unroll
    for (int i = 0; i < MT; ++i) acc[i] = vzero;

    #pragma unroll 8
    for (int k0 = 0; k0 < K; k0 += 4) {
        const int kb = k0 + (half << 1);
        v2f b;
        b.x = ldsB[kb * 16 + l16];
        b.y = ldsB[kb * 16 + 16 + l16];
        #pragma unroll
        for (int i = 0; i < MT; ++i) {
            // contiguous 8B load; immediate offset i*16*K*4 (<= 64 KB)
            v2f a = *(const v2f*)(a0 + i * 16 * K + k0);
            acc[i] = __builtin_amdgcn_wmma_f32_16x16x4_f32(
                /*neg_a=*/false, a, /*neg_b=*/false, b,
                /*c_mod=*/(short)0, acc[i], /*reuse_a=*/false, /*reuse_b=*/false);
        }
    }

    #pragma unroll
    for (int i = 0; i < MT; ++i) {
        const int mbase = (tm0 + i) * 16 + (half << 3);
        #pragma unroll
        for (int r = 0; r < 8; ++r)
            C[(size_t)(mbase + r) * NW + col] = acc[i][r];
    }
}

// ---------------- edge scatter (segment-sum with atomics) ------------------
// Virtual edges e in [E, E+Nn) are the self loops (s = d = e - E).
// One thread per (edge, float4 chunk); lc = log2(F/4).
__global__ void scatter_kernel(const int* __restrict__ src,
                               const int* __restrict__ dst,
                               const float* __restrict__ dinv,
                               const float* __restrict__ T,
                               float* __restrict__ acc,
                               int E, int Nn, int F, int lc) {
    const long long total = (long long)(E + Nn) << lc;
    long long idx = (long long)blockIdx.x * blockDim.x + threadIdx.x;
    if (idx >= total) return;

    const int e = (int)(idx >> lc);
    const int c = (int)(idx & ((1 << lc) - 1));

    int s, d;
    if (e < E) { s = src[e]; d = dst[e]; }
    else       { s = d = e - E; }

    const float w = dinv[s] * dinv[d];
    const float4 v = *(const float4*)(T + (size_t)s * F + (c << 2));
    float* p = acc + (size_t)d * F + (c << 2);
    atomicAdd(p + 0, w * v.x);
    atomicAdd(p + 1, w * v.y);
    atomicAdd(p + 2, w * v.z);
    atomicAdd(p + 3, w * v.w);
}

__global__ void bias_relu_kernel(float* __restrict__ h,
                                 const float* __restrict__ bias,
                                 long long total, int Fmask) {
    long long i = (long long)blockIdx.x * blockDim.x + threadIdx.x;
    if (i < total) {
        const int f = (int)i & Fmask;
        h[i] = fmaxf(h[i] + bias[f], 0.0f);
    }
}

// ---------------- inner-product decoder, replicated over K -----------------
// z rows are 16-stride (padded); only the first 8 features are live.
__global__ void decoder_kernel(const int* __restrict__ src,
                               const int* __restrict__ dst,
                               const float* __restrict__ z,   // [Nn x 16]
                               float* __restrict__ out,       // [K x E]
                               int E, int K) {
    int e = blockIdx.x * blockDim.x + threadIdx.x;
    if (e >= E) return;
    const int a = src[e], b = dst[e];
    const float4 x0 = *(const float4*)(z + (size_t)a * 16 + 0);
    const float4 x1 = *(const float4*)(z + (size_t)a * 16 + 4);
    const float4 y0 = *(const float4*)(z + (size_t)b * 16 + 0);
    const float4 y1 = *(const float4*)(z + (size_t)b * 16 + 4);
    const float v = x0.x * y0.x + x0.y * y0.y + x0.z * y0.z + x0.w * y0.w
                  + x1.x * y1.x + x1.y * y1.y + x1.z * y1.z + x1.w * y1.w;
    #pragma unroll
    for (int k = 0; k < 8; ++k)
        if (k < K) out[(size_t)k * E + e] = v;
}

// ---------------- host-side orchestration ----------------------------------

static inline void launch_fill(float* p, long long n, float v, hipStream_t s) {
    fill_f32_kernel<<<(unsigned)((n + 255) / 256), 256, 0, s>>>(p, n, v);
}

template <int K, int NW>
static inline void launch_gemm(const float* A, const float* B, float* C,
                               hipStream_t s) {
    gemm_wmma_kernel<K, NW><<<125 * (NW / 16), 160, 0, s>>>(A, B, C);
}

static inline void launch_scatter(const int* ei, const float* dinv,
                                  const float* T, float* acc,
                                  int E, int Nn, int F, int lc, hipStream_t s) {
    const long long total = (long long)(E + Nn) << lc;
    scatter_kernel<<<(unsigned)((total + 255) / 256), 256, 0, s>>>(
        ei, ei + E, dinv, T, acc, E, Nn, F, lc);
}

static inline void launch_bias_relu(float* h, const float* b, int Nn, int F,
                                    hipStream_t s) {
    const long long total = (long long)Nn * F;
    bias_relu_kernel<<<(unsigned)((total + 255) / 256), 256, 0, s>>>(
        h, b, total, F - 1);
}

extern "C" void kernel_launch(void* const* d_in, const int* in_sizes, int n_in,
                              void* d_out, int out_size, void* d_ws, size_t ws_size,
                              hipStream_t stream) {
    const float* x  = (const float*)d_in[0];
    const int*   ei = (const int*)  d_in[1];   // [2, E]: row0 = src, row1 = dst
    // d_in[2] = K scalar (fixed at 8 by the model definition)
    const float* W1 = (const float*)d_in[3];
    const float* b1 = (const float*)d_in[4];
    const float* W2 = (const float*)d_in[5];
    const float* b2 = (const float*)d_in[6];
    const float* W3 = (const float*)d_in[7];
    const float* b3 = (const float*)d_in[8];
    const float* Wk = (const float*)d_in[9];
    const float* bk = (const float*)d_in[10];
    float* out = (float*)d_out;

    const int Nn = in_sizes[0] / F_IN;   // 50000
    const int E  = in_sizes[1] / 2;      // 800000
    const int K  = out_size / E;         // 8

    // workspace layout (floats)
    float* ws   = (float*)d_ws;
    float* dinv = ws;                                    // [Nn]
    float* T    = ws + 50048;                            // [Nn x 256] pre-agg
    float* ACC  = T + (size_t)N_NODES * 256;             // [Nn x 256] aggregate
    float* Wkp  = ACC + (size_t)N_NODES * 256;           // [32 x 16]
    float* bkp  = Wkp + 32 * 16;                         // [16]

    // --- degree -> dinv (self loop contributes the initial 1.0) ---
    launch_fill(dinv, Nn, 1.0f, stream);
    deg_count_kernel<<<(E + 255) / 256, 256, 0, stream>>>(ei + E, dinv, E);
    rsqrt_kernel<<<(Nn + 255) / 256, 256, 0, stream>>>(dinv, Nn);
    pad_wk_kernel<<<1, 512, 0, stream>>>(Wk, bk, Wkp, bkp);

    // --- layer 1: 256 -> 256 ---
    launch_gemm<256, 256>(x, W1, T, stream);
    launch_fill(ACC, (long long)Nn * 256, 0.0f, stream);
    launch_scatter(ei, dinv, T, ACC, E, Nn, 256, 6, stream);
    launch_bias_relu(ACC, b1, Nn, 256, stream);          // h1 = ACC

    // --- layer 2: 256 -> 128 ---
    launch_gemm<256, 128>(ACC, W2, T, stream);
    launch_fill(ACC, (long long)Nn * 128, 0.0f, stream);
    launch_scatter(ei, dinv, T, ACC, E, Nn, 128, 5, stream);
    launch_bias_relu(ACC, b2, Nn, 128, stream);          // h2 = ACC

    // --- layer 3: 128 -> 32 ---
    launch_gemm<128, 32>(ACC, W3, T, stream);
    launch_fill(ACC, (long long)Nn * 32, 0.0f, stream);
    launch_scatter(ei, dinv, T, ACC, E, Nn, 32, 3, stream);
    launch_bias_relu(ACC, b3, Nn, 32, stream);           // h3 = ACC

    // --- K branch: 32 -> 8 (padded to 16; all K slices identical) ---
    launch_gemm<32, 16>(ACC, Wkp, T, stream);
    launch_fill(ACC, (long long)Nn * 16, 0.0f, stream);
    launch_scatter(ei, dinv, T, ACC, E, Nn, 16, 2, stream);
    launch_bias_relu(ACC, bkp, Nn, 16, stream);          // z = ACC (stride 16)

    // --- inner-product decoder, replicated over K ---
    decoder_kernel<<<(E + 255) / 256, 256, 0, stream>>>(ei, ei + E, ACC, out, E, K);
}